// TripletCenterLoss_81827716923629
// MI455X (gfx1250) — compile-verified
//
#include <hip/hip_runtime.h>
#include <math.h>

typedef float v2f __attribute__((ext_vector_type(2)));
typedef float v8f __attribute__((ext_vector_type(8)));

#define BATCH   4096
#define FEAT    512
#define NCLS    1000
#define NPAD    1024
#define MARGIN  0.3f
#define LAMBDA  0.01f
#define EPSD    1e-12f

// workspace layout (in floats)
#define WS_SQ     0        // 4096 floats: row squared norms
#define WS_AP     4096     // 4096 uints : hardest-positive dist bits (atomicMax)
#define WS_AN     8192     // 4096 uints : hardest-negative dist bits (atomicMin)
#define WS_ACC    12288    // 4 floats   : [0]=center_sum [1]=triplet_sum [2]=ce_sum
#define WS_LOGITS 16384    // 4096*1024 floats: logits (padded stride)

// ---------------------------------------------------------------- init
__global__ void init_ws_kernel(unsigned* __restrict__ apbits,
                               unsigned* __restrict__ anbits,
                               float* __restrict__ accum) {
    int i = blockIdx.x * blockDim.x + threadIdx.x;
    if (i < BATCH) {
        apbits[i] = 0u;            // dist >= 0, so 0 is identity for max
        anbits[i] = 0x7F800000u;   // +inf bits, identity for min (nonneg floats)
    }
    if (i < 4) accum[i] = 0.0f;
}

// ------------------------------------------------- row norms + center loss
__global__ __launch_bounds__(128)
void norms_center_kernel(const float* __restrict__ E, const int* __restrict__ lab,
                         const float* __restrict__ centers,
                         float* __restrict__ sq, float* __restrict__ accum) {
    const int row = blockIdx.x;
    const int t = threadIdx.x;
    const float* e = E + (size_t)row * FEAT;
    const float* c = centers + (size_t)lab[row] * FEAT;
    float s = 0.0f, cs = 0.0f;
    for (int f = t; f < FEAT; f += 128) {
        float v = e[f];
        s += v * v;
        float d = v - c[f];
        cs += d * d;
    }
    for (int o = 16; o; o >>= 1) {
        s  += __shfl_xor(s,  o, 32);
        cs += __shfl_xor(cs, o, 32);
    }
    __shared__ float sh[2][4];
    int wave = t >> 5, lane = t & 31;
    if (lane == 0) { sh[0][wave] = s; sh[1][wave] = cs; }
    __syncthreads();
    if (t == 0) {
        float S  = sh[0][0] + sh[0][1] + sh[0][2] + sh[0][3];
        float CS = sh[1][0] + sh[1][1] + sh[1][2] + sh[1][3];
        sq[row] = S;
        atomicAdd(&accum[0], CS);
    }
}

// ------------------------------- Gram via fp32 WMMA + hard pos/neg mining
// Block: 128 threads (4 waves). Tile: 128 rows x 64 cols.
// Each wave: 32x64 strip = 2x4 WMMA C-tiles (8 WMMAs per 6 LDS frag loads).
// Software pipeline: next K-slab prefetched into VGPRs during WMMA compute.
__global__ __launch_bounds__(128)
void gram_mine_kernel(const float* __restrict__ E, const int* __restrict__ lab,
                      const float* __restrict__ sq,
                      unsigned* __restrict__ apbits, unsigned* __restrict__ anbits) {
    __shared__ float As[128][33];  // pad to 33 floats/row -> conflict-free frag loads
    __shared__ float Bs[64][33];

    const int i0 = blockIdx.x * 128;
    const int j0 = blockIdx.y * 64;
    const int t = threadIdx.x;
    const int wave = t >> 5, lane = t & 31;
    const int half = lane >> 4, m = lane & 15;

    v8f acc[2][4] = {};

    const int c4 = t & 7;    // 4-float column group
    const int r0 = t >> 3;   // 0..15

    float4 pa[8], pb[4];
    #pragma unroll
    for (int rr = 0; rr < 8; ++rr)
        pa[rr] = *(const float4*)(E + (size_t)(i0 + r0 + rr*16) * FEAT + c4*4);
    #pragma unroll
    for (int rr = 0; rr < 4; ++rr)
        pb[rr] = *(const float4*)(E + (size_t)(j0 + r0 + rr*16) * FEAT + c4*4);

    for (int k0 = 0; k0 < FEAT; k0 += 32) {
        #pragma unroll
        for (int rr = 0; rr < 8; ++rr) {
            const int row = r0 + rr*16;
            As[row][c4*4+0] = pa[rr].x; As[row][c4*4+1] = pa[rr].y;
            As[row][c4*4+2] = pa[rr].z; As[row][c4*4+3] = pa[rr].w;
        }
        #pragma unroll
        for (int rr = 0; rr < 4; ++rr) {
            const int row = r0 + rr*16;
            Bs[row][c4*4+0] = pb[rr].x; Bs[row][c4*4+1] = pb[rr].y;
            Bs[row][c4*4+2] = pb[rr].z; Bs[row][c4*4+3] = pb[rr].w;
        }
        __syncthreads();

        if (k0 + 32 < FEAT) {  // prefetch next slab; waits land at next ds_store
            const int kn = k0 + 32;
            #pragma unroll
            for (int rr = 0; rr < 8; ++rr)
                pa[rr] = *(const float4*)(E + (size_t)(i0 + r0 + rr*16) * FEAT + kn + c4*4);
            #pragma unroll
            for (int rr = 0; rr < 4; ++rr)
                pb[rr] = *(const float4*)(E + (size_t)(j0 + r0 + rr*16) * FEAT + kn + c4*4);
        }

        #pragma unroll
        for (int kk = 0; kk < 32; kk += 4) {
            v2f a[2], b[4];
            #pragma unroll
            for (int rt = 0; rt < 2; ++rt) {
                a[rt].x = As[32*wave + 16*rt + m][kk + 2*half + 0];
                a[rt].y = As[32*wave + 16*rt + m][kk + 2*half + 1];
            }
            #pragma unroll
            for (int ct = 0; ct < 4; ++ct) {
                b[ct].x = Bs[16*ct + m][kk + 2*half + 0];
                b[ct].y = Bs[16*ct + m][kk + 2*half + 1];
            }
            #pragma unroll
            for (int rt = 0; rt < 2; ++rt)
                #pragma unroll
                for (int ct = 0; ct < 4; ++ct)
                    acc[rt][ct] = __builtin_amdgcn_wmma_f32_16x16x4_f32(
                        false, a[rt], false, b[ct], (short)0, acc[rt][ct], false, false);
        }
        __syncthreads();
    }

    // epilogue: Gram -> distance, hard mining (per 16-row tile rt)
    #pragma unroll
    for (int rt = 0; rt < 2; ++rt) {
        float sqi[8]; int labi[8];
        #pragma unroll
        for (int vr = 0; vr < 8; ++vr) {
            const int i = i0 + 32*wave + 16*rt + 8*half + vr;
            sqi[vr] = sq[i];
            labi[vr] = lab[i];
        }
        float mp[8], mn[8];
        #pragma unroll
        for (int vr = 0; vr < 8; ++vr) { mp[vr] = 0.0f; mn[vr] = INFINITY; }

        #pragma unroll
        for (int ct = 0; ct < 4; ++ct) {
            const int j = j0 + 16*ct + m;
            const float sqj = sq[j];
            const int labj = lab[j];
            #pragma unroll
            for (int vr = 0; vr < 8; ++vr) {
                float d2 = sqi[vr] + sqj - 2.0f * acc[rt][ct][vr];
                float dist = sqrtf(fmaxf(d2, EPSD));
                if (labi[vr] == labj) mp[vr] = fmaxf(mp[vr], dist);
                else                  mn[vr] = fminf(mn[vr], dist);
            }
        }
        #pragma unroll
        for (int vr = 0; vr < 8; ++vr) {
            float a = mp[vr], n = mn[vr];
            for (int s = 1; s < 16; s <<= 1) {
                a = fmaxf(a, __shfl_xor(a, s, 16));
                n = fminf(n, __shfl_xor(n, s, 16));
            }
            if (m == 0) {
                const int i = i0 + 32*wave + 16*rt + 8*half + vr;
                atomicMax(&apbits[i], __float_as_uint(a));
                atomicMin(&anbits[i], __float_as_uint(n));
            }
        }
    }
}

// ------------------------------------------ logits = E @ W^T + b via WMMA
__global__ __launch_bounds__(128)
void logits_gemm_kernel(const float* __restrict__ E, const float* __restrict__ W,
                        const float* __restrict__ bias, float* __restrict__ out) {
    __shared__ float As[128][33];
    __shared__ float Bs[64][33];

    const int i0 = blockIdx.x * 128;
    const int j0 = blockIdx.y * 64;
    const int t = threadIdx.x;
    const int wave = t >> 5, lane = t & 31;
    const int half = lane >> 4, m = lane & 15;

    v8f acc[2][4] = {};

    const int c4 = t & 7;
    const int r0 = t >> 3;

    float4 pa[8], pb[4];
    #pragma unroll
    for (int rr = 0; rr < 8; ++rr)
        pa[rr] = *(const float4*)(E + (size_t)(i0 + r0 + rr*16) * FEAT + c4*4);
    #pragma unroll
    for (int rr = 0; rr < 4; ++rr) {
        pb[rr] = make_float4(0.f, 0.f, 0.f, 0.f);
        if (j0 + r0 + rr*16 < NCLS)
            pb[rr] = *(const float4*)(W + (size_t)(j0 + r0 + rr*16) * FEAT + c4*4);
    }

    for (int k0 = 0; k0 < FEAT; k0 += 32) {
        #pragma unroll
        for (int rr = 0; rr < 8; ++rr) {
            const int row = r0 + rr*16;
            As[row][c4*4+0] = pa[rr].x; As[row][c4*4+1] = pa[rr].y;
            As[row][c4*4+2] = pa[rr].z; As[row][c4*4+3] = pa[rr].w;
        }
        #pragma unroll
        for (int rr = 0; rr < 4; ++rr) {
            const int row = r0 + rr*16;
            Bs[row][c4*4+0] = pb[rr].x; Bs[row][c4*4+1] = pb[rr].y;
            Bs[row][c4*4+2] = pb[rr].z; Bs[row][c4*4+3] = pb[rr].w;
        }
        __syncthreads();

        if (k0 + 32 < FEAT) {
            const int kn = k0 + 32;
            #pragma unroll
            for (int rr = 0; rr < 8; ++rr)
                pa[rr] = *(const float4*)(E + (size_t)(i0 + r0 + rr*16) * FEAT + kn + c4*4);
            #pragma unroll
            for (int rr = 0; rr < 4; ++rr) {
                pb[rr] = make_float4(0.f, 0.f, 0.f, 0.f);
                if (j0 + r0 + rr*16 < NCLS)
                    pb[rr] = *(const float4*)(W + (size_t)(j0 + r0 + rr*16) * FEAT + kn + c4*4);
            }
        }

        #pragma unroll
        for (int kk = 0; kk < 32; kk += 4) {
            v2f a[2], b[4];
            #pragma unroll
            for (int rt = 0; rt < 2; ++rt) {
                a[rt].x = As[32*wave + 16*rt + m][kk + 2*half + 0];
                a[rt].y = As[32*wave + 16*rt + m][kk + 2*half + 1];
            }
            #pragma unroll
            for (int ct = 0; ct < 4; ++ct) {
                b[ct].x = Bs[16*ct + m][kk + 2*half + 0];
                b[ct].y = Bs[16*ct + m][kk + 2*half + 1];
            }
            #pragma unroll
            for (int rt = 0; rt < 2; ++rt)
                #pragma unroll
                for (int ct = 0; ct < 4; ++ct)
                    acc[rt][ct] = __builtin_amdgcn_wmma_f32_16x16x4_f32(
                        false, a[rt], false, b[ct], (short)0, acc[rt][ct], false, false);
        }
        __syncthreads();
    }

    #pragma unroll
    for (int rt = 0; rt < 2; ++rt) {
        #pragma unroll
        for (int ct = 0; ct < 4; ++ct) {
            const int j = j0 + 16*ct + m;
            if (j < NCLS) {
                const float bj = bias[j];
                #pragma unroll
                for (int vr = 0; vr < 8; ++vr) {
                    const int i = i0 + 32*wave + 16*rt + 8*half + vr;
                    out[(size_t)i * NPAD + j] = acc[rt][ct][vr] + bj;
                }
            }
        }
    }
}

// ------------------------------------------------- cross entropy per row
__global__ __launch_bounds__(256)
void ce_rows_kernel(const float* __restrict__ logits, const int* __restrict__ lab,
                    float* __restrict__ accum) {
    const int row = blockIdx.x;
    const int t = threadIdx.x;
    const float* r = logits + (size_t)row * NPAD;

    float mx = -INFINITY;
    for (int c = t; c < NCLS; c += 256) mx = fmaxf(mx, r[c]);
    for (int o = 16; o; o >>= 1) mx = fmaxf(mx, __shfl_xor(mx, o, 32));
    __shared__ float sh[8];
    const int wave = t >> 5, lane = t & 31;
    if (lane == 0) sh[wave] = mx;
    __syncthreads();
    float rm = sh[0];
    #pragma unroll
    for (int w = 1; w < 8; ++w) rm = fmaxf(rm, sh[w]);
    __syncthreads();

    float s = 0.0f;
    for (int c = t; c < NCLS; c += 256) s += expf(r[c] - rm);
    for (int o = 16; o; o >>= 1) s += __shfl_xor(s, o, 32);
    if (lane == 0) sh[wave] = s;
    __syncthreads();
    if (t == 0) {
        float S = 0.0f;
        #pragma unroll
        for (int w = 0; w < 8; ++w) S += sh[w];
        const float lse = rm + logf(S);
        const float ce = lse - r[lab[row]];
        atomicAdd(&accum[2], ce);
    }
}

// ------------------------------------------------- triplet hinge reduce
__global__ __launch_bounds__(256)
void triplet_reduce_kernel(const unsigned* __restrict__ apbits,
                           const unsigned* __restrict__ anbits,
                           float* __restrict__ accum) {
    const int i = blockIdx.x * blockDim.x + threadIdx.x;
    float v = 0.0f;
    if (i < BATCH) {
        const float ap = __uint_as_float(apbits[i]);
        const float an = __uint_as_float(anbits[i]);
        v = fmaxf(ap - an + MARGIN, 0.0f);
    }
    for (int o = 16; o; o >>= 1) v += __shfl_xor(v, o, 32);
    __shared__ float sh[8];
    const int wave = threadIdx.x >> 5, lane = threadIdx.x & 31;
    if (lane == 0) sh[wave] = v;
    __syncthreads();
    if (threadIdx.x == 0) {
        float S = 0.0f;
        #pragma unroll
        for (int w = 0; w < 8; ++w) S += sh[w];
        atomicAdd(&accum[1], S);
    }
}

// --------------------------------------------------------------- finalize
__global__ void finalize_kernel(const float* __restrict__ accum, float* __restrict__ out) {
    if (blockIdx.x == 0 && threadIdx.x == 0) {
        const float invB = 1.0f / (float)BATCH;
        out[0] = accum[1] * invB + LAMBDA * (accum[0] * invB) + accum[2] * invB;
    }
}

// ---------------------------------------------------------------- launch
extern "C" void kernel_launch(void* const* d_in, const int* in_sizes, int n_in,
                              void* d_out, int out_size, void* d_ws, size_t ws_size,
                              hipStream_t stream) {
    const float* E       = (const float*)d_in[0];
    const int*   lab     = (const int*)d_in[1];
    const float* centers = (const float*)d_in[2];
    const float* W       = (const float*)d_in[3];
    const float* bias    = (const float*)d_in[4];
    float* out = (float*)d_out;

    float* w = (float*)d_ws;
    float*    sq     = w + WS_SQ;
    unsigned* apbits = (unsigned*)(w + WS_AP);
    unsigned* anbits = (unsigned*)(w + WS_AN);
    float*    accum  = w + WS_ACC;
    float*    logits = w + WS_LOGITS;

    init_ws_kernel<<<dim3(16), dim3(256), 0, stream>>>(apbits, anbits, accum);
    norms_center_kernel<<<dim3(BATCH), dim3(128), 0, stream>>>(E, lab, centers, sq, accum);
    gram_mine_kernel<<<dim3(32, 64), dim3(128), 0, stream>>>(E, lab, sq, apbits, anbits);
    logits_gemm_kernel<<<dim3(32, 16), dim3(128), 0, stream>>>(E, W, bias, logits);
    ce_rows_kernel<<<dim3(BATCH), dim3(256), 0, stream>>>(logits, lab, accum);
    triplet_reduce_kernel<<<dim3(16), dim3(256), 0, stream>>>(apbits, anbits, accum);
    finalize_kernel<<<dim3(1), dim3(1), 0, stream>>>(accum, out);
}